// CausalConv1d_67276367725129
// MI455X (gfx1250) — compile-verified
//
#include <hip/hip_runtime.h>
#include <stdint.h>

#define NSEQ   8
#define SEQLEN 2048
#define DCH    4096
#define KTAP   4
#define CHUNK  128
#define NCHUNK (SEQLEN / CHUNK)   /* 16 */
#define DVEC4  (DCH / 4)          /* 1024 float4 slots per row */
#define TPB    256
#define BLKS_PER_SC (DVEC4 / TPB) /* 4 blocks cover one (seq, chunk) */
#define NSTAGE 8                  /* async prefetch depth (rows in flight) */

typedef float v4f __attribute__((ext_vector_type(4)));
typedef int   v4i __attribute__((vector_size(16)));
typedef __attribute__((address_space(1))) v4i gv4i;   // global-AS vector
typedef __attribute__((address_space(3))) v4i lv4i;   // LDS-AS vector

__device__ __forceinline__ v4f nt_load(const float* p) {
    return __builtin_nontemporal_load((const v4f*)p);
}
__device__ __forceinline__ void nt_store(float* p, v4f v) {
    __builtin_nontemporal_store(v, (v4f*)p);
}
__device__ __forceinline__ float silu1(float v) {
    float e = __expf(-v);
    return v * __builtin_amdgcn_rcpf(1.0f + e);
}

// ---- CDNA5 async global->LDS copy (16B per lane), tracked by ASYNCcnt ----
__device__ __forceinline__ void async_load_b128(const float* g, float* l) {
#if __has_builtin(__builtin_amdgcn_global_load_async_to_lds_b128)
    __builtin_amdgcn_global_load_async_to_lds_b128(
        (gv4i*)const_cast<float*>(g), (lv4i*)l, 0, 0);
#else
    unsigned lds32 = (unsigned)(uintptr_t)l;            // low 32 bits = LDS byte addr
    unsigned long long g64 = (unsigned long long)(uintptr_t)g;
    asm volatile("global_load_async_to_lds_b128 %0, %1, off"
                 :: "v"(lds32), "v"(g64) : "memory");
#endif
}

template<int N>
__device__ __forceinline__ void wait_asynccnt() {
#if __has_builtin(__builtin_amdgcn_s_wait_asynccnt)
    __builtin_amdgcn_s_wait_asynccnt(N);
#else
    asm volatile("s_wait_asynccnt %0" :: "n"(N) : "memory");
#endif
}

__global__ __launch_bounds__(TPB)
void conv1d_stream_kernel(const float* __restrict__ x,
                          const float* __restrict__ weight,
                          const float* __restrict__ bias,
                          const float* __restrict__ conv_state,
                          const int*   __restrict__ state_ids,
                          const unsigned char* __restrict__ has_init,
                          float* __restrict__ out)
{
    __shared__ __align__(16) float stage_buf[NSTAGE * TPB * 4];  // 32 KB

    const int bid   = blockIdx.x;
    const int tid   = threadIdx.x;
    const int dvec  = (bid % BLKS_PER_SC) * TPB + tid;           // 0..1023
    const int chunk = (bid / BLKS_PER_SC) % NCHUNK;              // 0..15
    const int seq   =  bid / (BLKS_PER_SC * NCHUNK);             // 0..7
    const int d0    = dvec * 4;
    const int t0    = chunk * CHUNK;
    const long long rowbase = (long long)seq * SEQLEN;

    // ---- weights: (D,1,K) row-major -> 16 contiguous floats for 4 channels
    const float* wp = weight + (long long)d0 * KTAP;
    v4f wr0 = *(const v4f*)(wp + 0);
    v4f wr1 = *(const v4f*)(wp + 4);
    v4f wr2 = *(const v4f*)(wp + 8);
    v4f wr3 = *(const v4f*)(wp + 12);
    v4f w0 = { wr0.x, wr1.x, wr2.x, wr3.x };
    v4f w1 = { wr0.y, wr1.y, wr2.y, wr3.y };
    v4f w2 = { wr0.z, wr1.z, wr2.z, wr3.z };
    v4f w3 = { wr0.w, wr1.w, wr2.w, wr3.w };
    v4f bv = *(const v4f*)(bias + d0);

    // ---- prologue: issue NSTAGE async row prefetches (lane-private 16B slots)
    const float* gsrc = x + (rowbase + t0) * (long long)DCH + d0;
    float* myslot = stage_buf + tid * 4;   // this lane's slot inside a stage
#pragma unroll
    for (int s = 0; s < NSTAGE; ++s)
        async_load_b128(gsrc + (long long)s * DCH, myslot + s * (TPB * 4));

    // ---- tap history (x[t0-3..t0-1] or conv_state / zeros)
    v4f h0, h1, h2;
    if (t0 == 0) {
        if (has_init[seq]) {
            const int sid = state_ids[seq];
            const float* cs = conv_state + ((long long)sid * KTAP + 1) * DCH + d0;
            h0 = *(const v4f*)(cs);
            h1 = *(const v4f*)(cs + DCH);
            h2 = *(const v4f*)(cs + 2 * DCH);
        } else {
            h0 = (v4f)0.0f; h1 = (v4f)0.0f; h2 = (v4f)0.0f;
        }
    } else {
        const float* xp = x + (rowbase + t0 - 3) * (long long)DCH + d0;
        h0 = nt_load(xp);
        h1 = nt_load(xp + DCH);
        h2 = nt_load(xp + 2 * DCH);
    }

    float* oop = out + (rowbase + t0) * (long long)DCH + d0;
    const float* gnext = gsrc + (long long)NSTAGE * DCH;
    int stage = 0;

    // ---- steady state: wait oldest row, consume from LDS, refill its stage
#pragma unroll 2
    for (int t = 0; t < CHUNK - NSTAGE; ++t) {
        wait_asynccnt<NSTAGE - 1>();
        v4f xv  = *(const v4f*)(myslot + stage * (TPB * 4));   // ds_load_b128
        v4f acc = bv + w0 * h0 + w1 * h1 + w2 * h2 + w3 * xv;
        v4f r;
        r.x = silu1(acc.x);
        r.y = silu1(acc.y);
        r.z = silu1(acc.z);
        r.w = silu1(acc.w);
        nt_store(oop, r);
        h0 = h1; h1 = h2; h2 = xv;
        async_load_b128(gnext, myslot + stage * (TPB * 4));
        gnext += DCH;
        oop   += DCH;
        stage = (stage + 1) & (NSTAGE - 1);
    }

    // ---- drain: everything outstanding has landed after one wait
    wait_asynccnt<0>();
#pragma unroll
    for (int t = 0; t < NSTAGE; ++t) {
        v4f xv  = *(const v4f*)(myslot + stage * (TPB * 4));
        v4f acc = bv + w0 * h0 + w1 * h1 + w2 * h2 + w3 * xv;
        v4f r;
        r.x = silu1(acc.x);
        r.y = silu1(acc.y);
        r.z = silu1(acc.z);
        r.w = silu1(acc.w);
        nt_store(oop, r);
        h0 = h1; h1 = h2; h2 = xv;
        oop += DCH;
        stage = (stage + 1) & (NSTAGE - 1);
    }
}

// new_state[s, kk, d] = x[s*L + (L-4+kk), d], scattered to row state_ids[s]
__global__ __launch_bounds__(256)
void conv1d_state_kernel(const float* __restrict__ x,
                         const int*   __restrict__ state_ids,
                         float* __restrict__ out_state)
{
    const int idx  = blockIdx.x * blockDim.x + threadIdx.x;
    const int dvec = idx % DVEC4;
    const int kk   = (idx / DVEC4) % KTAP;
    const int s    = idx / (DVEC4 * KTAP);
    if (s >= NSEQ) return;
    const int sid = state_ids[s];
    const long long srow = (long long)s * SEQLEN + (SEQLEN - KTAP + kk);
    v4f v = *(const v4f*)(x + srow * DCH + dvec * 4);
    *(v4f*)(out_state + ((long long)sid * KTAP + kk) * DCH + dvec * 4) = v;
}

extern "C" void kernel_launch(void* const* d_in, const int* in_sizes, int n_in,
                              void* d_out, int out_size, void* d_ws, size_t ws_size,
                              hipStream_t stream) {
    (void)in_sizes; (void)n_in; (void)out_size; (void)d_ws; (void)ws_size;
    const float* x          = (const float*)d_in[0];
    const float* weight     = (const float*)d_in[1];
    const float* bias       = (const float*)d_in[2];
    const float* conv_state = (const float*)d_in[3];
    const int*   state_ids  = (const int*)d_in[4];
    const unsigned char* has_init = (const unsigned char*)d_in[5];

    float* out       = (float*)d_out;
    float* out_state = out + (long long)NSEQ * SEQLEN * DCH;

    (void)hipMemcpyAsync(out_state, conv_state,
                         (size_t)NSEQ * KTAP * DCH * sizeof(float),
                         hipMemcpyDeviceToDevice, stream);

    dim3 grid(NSEQ * NCHUNK * BLKS_PER_SC);   // 512 blocks x 8 wave32 = 4096 waves
    conv1d_stream_kernel<<<grid, TPB, 0, stream>>>(
        x, weight, bias, conv_state, state_ids, has_init, out);

    const int nstate_threads = NSEQ * KTAP * DVEC4;  // 32768
    conv1d_state_kernel<<<(nstate_threads + 255) / 256, 256, 0, stream>>>(
        x, state_ids, out_state);
}